// TransitionDown_65042984730710
// MI455X (gfx1250) — compile-verified
//
#include <hip/hip_runtime.h>
#include <stdint.h>

// ---------------- problem constants ----------------
#define N_PTS 60000
#define M_PTS 15000
#define KNN_K 16
#define CIN   64
#define CH    67          // 3 + CIN
#define CHP   68          // padded contraction dim (multiple of 4)
#define GROW  72          // LDS row stride for g tile
#define COUT  128
#define BN_EPS 1e-5f

// ---------------- FPS config ----------------
#define FPS_BLOCKS 32
#define FPS_THREADS 256
#define PTS_PER_BLOCK (N_PTS / FPS_BLOCKS)   // 1875

// ---------------- workspace layout (bytes) ----------------
#define WS_CNT    0            // u64 barrier counter
#define WS_CAND   64           // u64 cand[2][FPS_BLOCKS]
#define WS_IDX    1024         // int idx_fps[M_PTS]
#define WS_NN     65536        // int nn_idx[M_PTS*KNN_K]
#define WS_WT     1048576      // float Wt[CHP][COUT]
#define WS_SUMS   2097152      // float sums[COUT]
#define WS_SUMSQ  (WS_SUMS + 512)
#define WS_SCALE  (WS_SUMS + 1024)
#define WS_SHIFT  (WS_SUMS + 1536)

// ---------------- d_out layout (floats) ----------------
#define NP_OFF  0
#define OUT_OFF (M_PTS * 3)
#define RS_OFF  (OUT_OFF + M_PTS * COUT)

typedef float v2f __attribute__((ext_vector_type(2)));
typedef float v8f __attribute__((ext_vector_type(8)));

typedef unsigned long long u64;
typedef unsigned int u32;

__device__ __forceinline__ u64 shfl_xor_u64(u64 v, int m) {
  u32 lo = (u32)v, hi = (u32)(v >> 32);
  lo = __shfl_xor(lo, m, 32);
  hi = __shfl_xor(hi, m, 32);
  return ((u64)hi << 32) | (u64)lo;
}

// ================= init: zero barrier counter + BN accumulators =================
__global__ void init_kernel(u64* cnt, float* sums, float* sumsq) {
  int t = threadIdx.x;
  if (t == 0) *cnt = 0ull;
  if (t < COUT) { sums[t] = 0.f; sumsq[t] = 0.f; }
}

// ================= FPS: 32 persistent blocks, LDS-resident slices =================
__global__ __launch_bounds__(FPS_THREADS)
void fps_kernel(const float* __restrict__ point, int* __restrict__ idx_fps,
                u64* cnt, u64* cand) {
  __shared__ float lx[PTS_PER_BLOCK], ly[PTS_PER_BLOCK], lz[PTS_PER_BLOCK], ld[PTS_PER_BLOCK];
  __shared__ u64 wkey[FPS_THREADS / 32];
  __shared__ int s_win;

  const int tid = threadIdx.x;
  const int bid = blockIdx.x;
  const int base = bid * PTS_PER_BLOCK;

  for (int i = tid; i < PTS_PER_BLOCK; i += FPS_THREADS) {
    lx[i] = point[(size_t)(base + i) * 3 + 0];
    ly[i] = point[(size_t)(base + i) * 3 + 1];
    lz[i] = point[(size_t)(base + i) * 3 + 2];
    ld[i] = __builtin_inff();
  }
  if (bid == 0 && tid == 0) idx_fps[0] = 0;
  __syncthreads();

  int last = 0;
  for (int step = 1; step < M_PTS; ++step) {
    const float cx = point[(size_t)last * 3 + 0];
    const float cy = point[(size_t)last * 3 + 1];
    const float cz = point[(size_t)last * 3 + 2];

    u64 key = 0ull;
    for (int i = tid; i < PTS_PER_BLOCK; i += FPS_THREADS) {
      float dx = lx[i] - cx, dy = ly[i] - cy, dz = lz[i] - cz;
      float d = dx * dx + dy * dy + dz * dz;
      float cur = ld[i];
      if (d < cur) { cur = d; ld[i] = d; }
      u64 k = ((u64)__float_as_uint(cur) << 32) | (u32)(~(u32)(base + i));
      key = (k > key) ? k : key;
    }
    for (int off = 16; off; off >>= 1) {
      u64 o = shfl_xor_u64(key, off);
      if (o > key) key = o;
    }
    if ((tid & 31) == 0) wkey[tid >> 5] = key;
    __syncthreads();

    if (tid == 0) {
      u64 bk = wkey[0];
      for (int w = 1; w < FPS_THREADS / 32; ++w) if (wkey[w] > bk) bk = wkey[w];
      const int buf = step & 1;
      __hip_atomic_store(&cand[buf * FPS_BLOCKS + bid], bk,
                         __ATOMIC_RELEASE, __HIP_MEMORY_SCOPE_AGENT);
      __hip_atomic_fetch_add(cnt, 1ull, __ATOMIC_ACQ_REL, __HIP_MEMORY_SCOPE_AGENT);
      const u64 target = (u64)FPS_BLOCKS * (u64)step;
      while (__hip_atomic_load(cnt, __ATOMIC_ACQUIRE, __HIP_MEMORY_SCOPE_AGENT) < target)
        __builtin_amdgcn_s_sleep(1);
    }
    __syncthreads();

    if (tid < 32) {
      const int buf = step & 1;
      u64 k = __hip_atomic_load(&cand[buf * FPS_BLOCKS + tid],
                                __ATOMIC_RELAXED, __HIP_MEMORY_SCOPE_AGENT);
      for (int off = 16; off; off >>= 1) {
        u64 o = shfl_xor_u64(k, off);
        if (o > k) k = o;
      }
      if (tid == 0) {
        int win = (int)(~(u32)(k & 0xFFFFFFFFull));
        s_win = win;
        if (bid == 0) idx_fps[step] = win;
      }
    }
    __syncthreads();
    last = s_win;
  }
}

// ================= gather sampled points + row_splits =================
__global__ void newpoint_kernel(const float* __restrict__ point,
                                const int* __restrict__ idx_fps,
                                float* __restrict__ dout) {
  int t = blockIdx.x * blockDim.x + threadIdx.x;
  if (t < M_PTS) {
    int s = idx_fps[t];
    dout[NP_OFF + t * 3 + 0] = point[(size_t)s * 3 + 0];
    dout[NP_OFF + t * 3 + 1] = point[(size_t)s * 3 + 1];
    dout[NP_OFF + t * 3 + 2] = point[(size_t)s * 3 + 2];
  }
  if (t == 0) {
    int* rs = (int*)(dout + RS_OFF);
    rs[0] = 0; rs[1] = M_PTS;
  }
}

// ================= KNN: thread per query, LDS point tiles, top-16 in regs ========
#define KNN_TILE 512
__global__ __launch_bounds__(256)
void knn_kernel(const float* __restrict__ point, const float* __restrict__ npnt,
                int* __restrict__ nn) {
  __shared__ float tx[KNN_TILE], ty[KNN_TILE], tz[KNN_TILE];
  const int tid = threadIdx.x;
  const int q = blockIdx.x * 256 + tid;
  const bool valid = q < M_PTS;
  float qx = 0.f, qy = 0.f, qz = 0.f;
  if (valid) { qx = npnt[q * 3 + 0]; qy = npnt[q * 3 + 1]; qz = npnt[q * 3 + 2]; }

  float bd[KNN_K]; int bi[KNN_K];
#pragma unroll
  for (int i = 0; i < KNN_K; ++i) { bd[i] = __builtin_inff(); bi[i] = 0; }

  for (int basep = 0; basep < N_PTS; basep += KNN_TILE) {
    const int cntp = min(KNN_TILE, N_PTS - basep);
    __syncthreads();
    for (int j = tid; j < cntp; j += 256) {
      tx[j] = point[(size_t)(basep + j) * 3 + 0];
      ty[j] = point[(size_t)(basep + j) * 3 + 1];
      tz[j] = point[(size_t)(basep + j) * 3 + 2];
    }
    __syncthreads();
    if (valid) {
      for (int j = 0; j < cntp; ++j) {
        float dx = tx[j] - qx, dy = ty[j] - qy, dz = tz[j] - qz;
        float d = dx * dx + dy * dy + dz * dz;
        if (d < bd[KNN_K - 1]) {
          int pos = KNN_K - 1;
#pragma unroll
          for (int s = KNN_K - 2; s >= 0; --s) {
            if (bd[s] > d) { bd[s + 1] = bd[s]; bi[s + 1] = bi[s]; pos = s; }
          }
#pragma unroll
          for (int s = 0; s < KNN_K; ++s) {
            if (s == pos) { bd[s] = d; bi[s] = basep + j; }
          }
        }
      }
    }
  }
  if (valid) {
#pragma unroll
    for (int i = 0; i < KNN_K; ++i) nn[q * KNN_K + i] = bi[i];
  }
}

// ================= transpose/pad W -> Wt[CHP][COUT] =================
__global__ void wt_kernel(const float* __restrict__ W, float* __restrict__ Wt) {
  int t = blockIdx.x * blockDim.x + threadIdx.x;
  if (t < CHP * COUT) {
    int c = t / COUT;   // channel (contraction) index
    int o = t % COUT;   // output channel
    Wt[t] = (c < CH) ? W[o * CH + c] : 0.f;
  }
}

// ================= shared WMMA helpers =================
// g tile: 16 rows (k-neighbors) x 68 channels, LDS, row stride GROW
__device__ __forceinline__ void build_g(float* gw, int m, int lane,
                                        const float* __restrict__ point,
                                        const float* __restrict__ feat,
                                        const float* __restrict__ npnt,
                                        const int* __restrict__ nn) {
  const int r = lane & 15;
  const int half = lane >> 4;
  const int nidx = nn[m * KNN_K + r];
  const float* prow = point + (size_t)nidx * 3;
  const float* frow = feat + (size_t)nidx * CIN;
  const float* q = npnt + (size_t)m * 3;
  float* grow = gw + r * GROW;
  if (half == 0) {
    grow[0] = prow[0] - q[0];
    grow[1] = prow[1] - q[1];
    grow[2] = prow[2] - q[2];
    for (int t = 0; t < 31; ++t) grow[3 + t] = frow[t];       // channels 3..33
  } else {
    for (int t = 0; t < 33; ++t) grow[34 + t] = frow[31 + t]; // channels 34..66
    grow[67] = 0.f;                                           // pad
  }
}

// one 16x16 f32 tile via V_WMMA_F32_16X16X4_F32, contraction over CHP
__device__ __forceinline__ v8f wmma_tile(const float* gw, const float* __restrict__ Wt,
                                         int ct, int lane) {
  const int r = lane & 15;      // A: M row ; B/C/D: N column
  const int half = lane >> 4;
  v8f acc = {0.f, 0.f, 0.f, 0.f, 0.f, 0.f, 0.f, 0.f};
  const float* arow = gw + r * GROW + half * 2;
  const float* bcol = Wt + ct * 16 + r;
#pragma unroll
  for (int c0 = 0; c0 < CHP; c0 += 4) {
    v2f a; a.x = arow[c0]; a.y = arow[c0 + 1];
    const int kk = c0 + half * 2;
    v2f b; b.x = bcol[kk * COUT]; b.y = bcol[(kk + 1) * COUT];
    acc = __builtin_amdgcn_wmma_f32_16x16x4_f32(false, a, false, b,
                                                (short)0, acc, false, false);
  }
  return acc;
}

// ================= pass A: WMMA GEMM + per-channel sum / sumsq =================
__global__ __launch_bounds__(256)
void passA_kernel(const float* __restrict__ point, const float* __restrict__ feat,
                  const float* __restrict__ npnt, const int* __restrict__ nn,
                  const float* __restrict__ Wt,
                  float* __restrict__ sums, float* __restrict__ sumsq) {
  __shared__ float gtile[8][16 * GROW];
  __shared__ float ls[COUT], lsq[COUT];
  const int tid = threadIdx.x, wid = tid >> 5, lane = tid & 31;
  if (tid < COUT) { ls[tid] = 0.f; lsq[tid] = 0.f; }
  const int m = blockIdx.x * 8 + wid;
  build_g(gtile[wid], m, lane, point, feat, npnt, nn);
  __syncthreads();
  for (int ct = 0; ct < COUT / 16; ++ct) {
    v8f acc = wmma_tile(gtile[wid], Wt, ct, lane);
    float s = 0.f, q = 0.f;
#pragma unroll
    for (int v = 0; v < 8; ++v) { float y = acc[v]; s += y; q += y * y; }
    s += __shfl_xor(s, 16, 32);
    q += __shfl_xor(q, 16, 32);
    if ((lane >> 4) == 0) {
      atomicAdd(&ls[ct * 16 + lane], s);
      atomicAdd(&lsq[ct * 16 + lane], q);
    }
  }
  __syncthreads();
  if (tid < COUT) {
    atomicAdd(&sums[tid], ls[tid]);
    atomicAdd(&sumsq[tid], lsq[tid]);
  }
}

// ================= fold stats into BN scale/shift =================
__global__ void bnparam_kernel(const float* __restrict__ sums, const float* __restrict__ sumsq,
                               const float* __restrict__ gamma, const float* __restrict__ beta,
                               float* __restrict__ sc, float* __restrict__ sh) {
  int c = threadIdx.x;
  if (c < COUT) {
    const float inv = 1.0f / (float)(M_PTS * KNN_K);
    float mean = sums[c] * inv;
    float var = sumsq[c] * inv - mean * mean;
    float s = gamma[c] * rsqrtf(var + BN_EPS);
    sc[c] = s;
    sh[c] = beta[c] - mean * s;
  }
}

// ================= pass B: WMMA GEMM + BN + ReLU + max over K =================
__global__ __launch_bounds__(256)
void passB_kernel(const float* __restrict__ point, const float* __restrict__ feat,
                  float* __restrict__ dout, const int* __restrict__ nn,
                  const float* __restrict__ Wt,
                  const float* __restrict__ sc, const float* __restrict__ sh) {
  __shared__ float gtile[8][16 * GROW];
  const int tid = threadIdx.x, wid = tid >> 5, lane = tid & 31;
  const float* npnt = dout + NP_OFF;
  const int m = blockIdx.x * 8 + wid;
  build_g(gtile[wid], m, lane, point, feat, npnt, nn);
  __syncthreads();
  for (int ct = 0; ct < COUT / 16; ++ct) {
    v8f acc = wmma_tile(gtile[wid], Wt, ct, lane);
    const int ch = ct * 16 + (lane & 15);
    const float a = sc[ch], b = sh[ch];
    float mx = 0.f;  // ReLU floor
#pragma unroll
    for (int v = 0; v < 8; ++v) {
      float y = fmaxf(acc[v] * a + b, 0.f);
      mx = fmaxf(mx, y);
    }
    mx = fmaxf(mx, __shfl_xor(mx, 16, 32));
    if ((lane >> 4) == 0) dout[OUT_OFF + (size_t)m * COUT + ch] = mx;
  }
}

// ================= host launch =================
extern "C" void kernel_launch(void* const* d_in, const int* in_sizes, int n_in,
                              void* d_out, int out_size, void* d_ws, size_t ws_size,
                              hipStream_t stream) {
  const float* point = (const float*)d_in[0];
  const float* feat  = (const float*)d_in[1];
  // d_in[2] = row_splits (unused: single cloud)
  const float* W     = (const float*)d_in[3];
  const float* gamma = (const float*)d_in[4];
  const float* beta  = (const float*)d_in[5];
  float* out = (float*)d_out;

  char* ws = (char*)d_ws;
  u64*   cnt     = (u64*)(ws + WS_CNT);
  u64*   cand    = (u64*)(ws + WS_CAND);
  int*   idx_fps = (int*)(ws + WS_IDX);
  int*   nn      = (int*)(ws + WS_NN);
  float* Wt      = (float*)(ws + WS_WT);
  float* sums    = (float*)(ws + WS_SUMS);
  float* sumsq   = (float*)(ws + WS_SUMSQ);
  float* sc      = (float*)(ws + WS_SCALE);
  float* sh      = (float*)(ws + WS_SHIFT);

  hipLaunchKernelGGL(init_kernel, dim3(1), dim3(128), 0, stream, cnt, sums, sumsq);
  hipLaunchKernelGGL(fps_kernel, dim3(FPS_BLOCKS), dim3(FPS_THREADS), 0, stream,
                     point, idx_fps, cnt, cand);
  hipLaunchKernelGGL(newpoint_kernel, dim3((M_PTS + 255) / 256), dim3(256), 0, stream,
                     point, idx_fps, out);
  hipLaunchKernelGGL(knn_kernel, dim3((M_PTS + 255) / 256), dim3(256), 0, stream,
                     point, out + NP_OFF, nn);
  hipLaunchKernelGGL(wt_kernel, dim3((CHP * COUT + 255) / 256), dim3(256), 0, stream, W, Wt);
  hipLaunchKernelGGL(passA_kernel, dim3(M_PTS / 8), dim3(256), 0, stream,
                     point, feat, out + NP_OFF, nn, Wt, sums, sumsq);
  hipLaunchKernelGGL(bnparam_kernel, dim3(1), dim3(128), 0, stream,
                     sums, sumsq, gamma, beta, sc, sh);
  hipLaunchKernelGGL(passB_kernel, dim3(M_PTS / 8), dim3(256), 0, stream,
                     point, feat, out, nn, Wt, sc, sh);
}